// MultiRelGraphConv_23862838297343
// MI455X (gfx1250) — compile-verified
//
#include <hip/hip_runtime.h>

#define D 128
#define RRELU_SLOPE 0.22916666666666666f  // (1/8 + 1/3)/2

typedef float v2f __attribute__((ext_vector_type(2)));
typedef float v8f __attribute__((ext_vector_type(8)));

__device__ __forceinline__ v8f wmma_f32_k4(v2f a, v2f b, v8f c) {
  // D = A(16x4) * B(4x16) + C, fp32 WMMA (CDNA5)
  return __builtin_amdgcn_wmma_f32_16x16x4_f32(false, a, false, b, (short)0, c, false, false);
}

// Stage a 128x128 (K x N) weight chunk into LDS, K-interleaved by groups of 4:
//   ldsW[kk*512 + col*4 + j] = Wsrc[(kk*4 + j)*128 + col],  kk in [0,32), j in [0,4)
// This makes each WMMA B fragment (two consecutive K rows, same col) one
// contiguous 8B ds_load_b64, and the wave's 32 lanes cover all 64 LDS banks.
__device__ __forceinline__ void stage_w_chunk(const float* __restrict__ Wsrc,
                                              float* __restrict__ ldsW, int tid) {
  const int col = tid & 127;
  const int kk0 = tid >> 7;  // 0 or 1
  #pragma unroll
  for (int i = 0; i < 16; ++i) {
    const int kk = kk0 + i * 2;
    float4 v;
    v.x = Wsrc[(kk * 4 + 0) * D + col];
    v.y = Wsrc[(kk * 4 + 1) * D + col];
    v.z = Wsrc[(kk * 4 + 2) * D + col];
    v.w = Wsrc[(kk * 4 + 3) * D + col];
    *(float4*)(ldsW + kk * 512 + col * 4) = v;
  }
}

__global__ void zero_f32(float* __restrict__ p, long long n) {
  long long i = (long long)blockIdx.x * blockDim.x + threadIdx.x;
  long long stride = (long long)gridDim.x * blockDim.x;
  for (; i < n; i += stride) p[i] = 0.0f;
}

__global__ void deg_count(const int* __restrict__ dst, float* __restrict__ deg, int n_edges) {
  int i = blockIdx.x * blockDim.x + threadIdx.x;
  if (i < n_edges) atomicAdd(&deg[dst[i]], 1.0f);
}

__global__ void make_rdeg(float* __restrict__ deg, int n) {
  int i = blockIdx.x * blockDim.x + threadIdx.x;
  if (i < n) deg[i] = 1.0f / fmaxf(deg[i], 1.0f);
}

// Per-edge message GEMM + atomic scatter:
//   msg = [h[src] | edge_feats] @ W1 + b1 ; agg[dst] += msg
// One wave = 16 edges x 128 cols. K=256 done as two K=128 halves with W1 staged in LDS.
__global__ __launch_bounds__(256) void edge_msg(
    const float* __restrict__ h, const float* __restrict__ ef,
    const int* __restrict__ src, const int* __restrict__ dst,
    const float* __restrict__ W1, const float* __restrict__ b1,
    float* __restrict__ agg, int n_edges)
{
  __shared__ __align__(16) float ldsW[D * D];  // 64KB: one K-half of W1, interleaved
  const int tid  = threadIdx.x;
  const int wave = tid >> 5;
  const int lane = tid & 31;
  const int row  = lane & 15;
  const int koff = (lane >> 4) << 1;  // lanes 16..31 carry K+2,K+3

  const long long ebase = ((long long)blockIdx.x * 8 + wave) * 16;
  long long eA = ebase + row;
  if (eA >= n_edges) eA = n_edges - 1;          // clamp (masked at scatter)
  const int   sA   = src[eA];
  const float* __restrict__ hrow = h  + (size_t)sA * D;
  const float* __restrict__ erow = ef + (size_t)eA * D;

  v8f acc[8] = {};

  for (int half = 0; half < 2; ++half) {
    __syncthreads();
    stage_w_chunk(W1 + (size_t)half * D * D, ldsW, tid);
    __syncthreads();

    const float* __restrict__ arow = (half == 0) ? hrow : erow;
    for (int kk = 0; kk < 32; ++kk) {
      v2f a = *(const v2f*)(arow + kk * 4 + koff);
      const float* __restrict__ wb = ldsW + kk * 512 + row * 4 + koff;
      #pragma unroll
      for (int n = 0; n < 8; ++n) {
        v2f b = *(const v2f*)(wb + n * 64);
        acc[n] = wmma_f32_k4(a, b, acc[n]);
      }
    }
  }

  // C layout: acc[n][r] = C[M = r + 8*hi][N = 16n + row]
  const int hi8 = (lane >> 4) * 8;
  #pragma unroll
  for (int r = 0; r < 8; ++r) {
    const long long e = ebase + hi8 + r;
    if (e < n_edges) {
      float* __restrict__ arow = agg + (size_t)dst[e] * D;
      #pragma unroll
      for (int n = 0; n < 8; ++n) {
        const int col = n * 16 + row;
        atomicAdd(&arow[col], acc[n][r] + b1[col]);
      }
    }
  }
}

// h_out = lrelu(a + a@W2 + b2), a = agg * rdeg
__global__ __launch_bounds__(256) void node_update(
    const float* __restrict__ agg, const float* __restrict__ rdeg,
    const float* __restrict__ W2, const float* __restrict__ b2,
    float* __restrict__ hout, int n_nodes)
{
  __shared__ __align__(16) float ldsW[D * D];
  const int tid = threadIdx.x, wave = tid >> 5, lane = tid & 31;
  const int row = lane & 15, koff = (lane >> 4) << 1;

  stage_w_chunk(W2, ldsW, tid);
  __syncthreads();

  const long long vbase = ((long long)blockIdx.x * 8 + wave) * 16;
  long long vA = vbase + row;
  if (vA >= n_nodes) vA = n_nodes - 1;
  const float rd = rdeg[vA];
  const float* __restrict__ arow = agg + (size_t)vA * D;

  v8f acc[8] = {};
  for (int kk = 0; kk < 32; ++kk) {
    v2f a = *(const v2f*)(arow + kk * 4 + koff);
    a.x *= rd; a.y *= rd;
    const float* __restrict__ wb = ldsW + kk * 512 + row * 4 + koff;
    #pragma unroll
    for (int n = 0; n < 8; ++n) {
      v2f b = *(const v2f*)(wb + n * 64);
      acc[n] = wmma_f32_k4(a, b, acc[n]);
    }
  }

  const int hi8 = (lane >> 4) * 8;
  #pragma unroll
  for (int r = 0; r < 8; ++r) {
    const long long v = vbase + hi8 + r;
    if (v < n_nodes) {
      const float rdv = rdeg[v];
      const float* __restrict__ ar = agg + (size_t)v * D;
      float* __restrict__ hr = hout + (size_t)v * D;
      #pragma unroll
      for (int n = 0; n < 8; ++n) {
        const int col = n * 16 + row;
        float x = ar[col] * rdv + acc[n][r] + b2[col];
        hr[col] = (x >= 0.0f) ? x : x * RRELU_SLOPE;
      }
    }
  }
}

// out = [x0 | x1 | x2] @ Wf + bf  (K = 384 in three 128-row chunks of Wf via LDS)
__global__ __launch_bounds__(256) void final_proj(
    const float* __restrict__ x0, const float* __restrict__ x1, const float* __restrict__ x2,
    const float* __restrict__ Wf, const float* __restrict__ bf,
    float* __restrict__ out, int n_nodes)
{
  __shared__ __align__(16) float ldsW[D * D];
  const int tid = threadIdx.x, wave = tid >> 5, lane = tid & 31;
  const int row = lane & 15, koff = (lane >> 4) << 1;

  const long long vbase = ((long long)blockIdx.x * 8 + wave) * 16;
  long long vA = vbase + row;
  if (vA >= n_nodes) vA = n_nodes - 1;

  const float* __restrict__ segs[3] = {
      x0 + (size_t)vA * D, x1 + (size_t)vA * D, x2 + (size_t)vA * D };

  v8f acc[8] = {};
  for (int seg = 0; seg < 3; ++seg) {
    __syncthreads();
    stage_w_chunk(Wf + (size_t)seg * D * D, ldsW, tid);
    __syncthreads();

    const float* __restrict__ arow = segs[seg];
    for (int kk = 0; kk < 32; ++kk) {
      v2f a = *(const v2f*)(arow + kk * 4 + koff);
      const float* __restrict__ wb = ldsW + kk * 512 + row * 4 + koff;
      #pragma unroll
      for (int n = 0; n < 8; ++n) {
        v2f b = *(const v2f*)(wb + n * 64);
        acc[n] = wmma_f32_k4(a, b, acc[n]);
      }
    }
  }

  const int hi8 = (lane >> 4) * 8;
  #pragma unroll
  for (int r = 0; r < 8; ++r) {
    const long long v = vbase + hi8 + r;
    if (v < n_nodes) {
      float* __restrict__ orow = out + (size_t)v * D;
      #pragma unroll
      for (int n = 0; n < 8; ++n) {
        const int col = n * 16 + row;
        orow[col] = acc[n][r] + bf[col];
      }
    }
  }
}

extern "C" void kernel_launch(void* const* d_in, const int* in_sizes, int n_in,
                              void* d_out, int out_size, void* d_ws, size_t ws_size,
                              hipStream_t stream) {
  const float* node_feats = (const float*)d_in[0];
  const float* edge_feats = (const float*)d_in[1];
  const int*   src        = (const int*)d_in[2];
  const int*   dst        = (const int*)d_in[3];
  const float* W1a[2] = {(const float*)d_in[4], (const float*)d_in[8]};
  const float* b1a[2] = {(const float*)d_in[5], (const float*)d_in[9]};
  const float* W2a[2] = {(const float*)d_in[6], (const float*)d_in[10]};
  const float* b2a[2] = {(const float*)d_in[7], (const float*)d_in[11]};
  const float* Wf = (const float*)d_in[12];
  const float* bf = (const float*)d_in[13];

  const int N = in_sizes[0] / D;
  const int E = in_sizes[2];

  size_t off = 0;
  auto wsalloc = [&](size_t bytes) -> float* {
    float* p = (float*)((char*)d_ws + off);
    off += (bytes + 255) & ~(size_t)255;
    return p;
  };
  float* deg = wsalloc((size_t)N * 4);
  float* agg = wsalloc((size_t)N * D * 4);
  float* h1  = wsalloc((size_t)N * D * 4);
  float* h2  = wsalloc((size_t)N * D * 4);

  dim3 blk(256);
  const int nodeBlocks = (N + 127) / 128;  // 8 waves x 16 rows per block
  const int edgeBlocks = (E + 127) / 128;

  zero_f32<<<512, blk, 0, stream>>>(deg, (long long)N);
  deg_count<<<(E + 255) / 256, blk, 0, stream>>>(dst, deg, E);
  make_rdeg<<<(N + 255) / 256, blk, 0, stream>>>(deg, N);

  const float* hcur = node_feats;
  float* houts[2] = {h1, h2};
  for (int l = 0; l < 2; ++l) {
    zero_f32<<<2048, blk, 0, stream>>>(agg, (long long)N * D);
    edge_msg<<<edgeBlocks, blk, 0, stream>>>(hcur, edge_feats, src, dst,
                                             W1a[l], b1a[l], agg, E);
    node_update<<<nodeBlocks, blk, 0, stream>>>(agg, deg, W2a[l], b2a[l], houts[l], N);
    hcur = houts[l];
  }
  final_proj<<<nodeBlocks, blk, 0, stream>>>(node_feats, h1, h2, Wf, bf,
                                             (float*)d_out, N);
}